// InferModel_12206297055551
// MI455X (gfx1250) — compile-verified
//
#include <hip/hip_runtime.h>
#include <hip/hip_bf16.h>

// ---------------------------------------------------------------------------
// CenterNet-style decode for MI455X (gfx1250, wave32).
//   out = (bboxes[16,64,6], seg_feat[16,8,128,128], conv_w[16,64,169])
// Memory-bound: ~102 MB of traffic -> ~4.4us floor @ 23.3 TB/s.
// CDNA5-specific paths used:
//   * TENSOR_LOAD_TO_LDS (TDM) with pad-insertion: one descriptor DMA loads
//     the whole 128x128 hm plane into a stride-136 padded LDS tile
//   * global_load_async_to_lds_b128 + s_wait_asynccnt (ASYNCcnt path) for
//     stage-2 candidate staging
//   * 320KB-per-WGP LDS: full padded plane + 16K key array resident
//   * wave32 LDS atomics (ds_add) for radix histograms / compaction
// No matmul exists in this op, so WMMA does not apply.
// ---------------------------------------------------------------------------

#define B_   16
#define C_   80
#define H_   128
#define W_   128
#define HW_  (H_ * W_)      // 16384
#define K_   64
#define CK_  (C_ * K_)      // 5120

// Padded LDS tile: 130 rows x 136 floats. Image row r lives at
// (r+1)*136 + 4 + c. TDM writes rows of 128 DWORDs then skips 8 DWORDs
// (pad_interval=6 -> 2^6 8-byte units = 128 DWORDs; pad_amount=7 -> 8 DWORDs),
// which lands row r exactly at byte 560 + r*544 = ((r+1)*136+4)*4.
#define PSTRIDE 136
#define PROWS   130
#define PFLOATS (PROWS * PSTRIDE)         // 17680
#define PBYTES  (PFLOATS * 4)             // 70720
#define KEYS_OFF   PBYTES                 // 16384 u32 keys
#define HIST_OFF   (PBYTES + HW_ * 4)     // 256 u32 histogram
#define CTRL_OFF   (HIST_OFF + 1024)      // a few control words
#define SMEM1_BYTES (CTRL_OFF + 64)       // 137,344 B (< 160KB -> 2 WG/WGP)

// Stage-2 dynamic LDS layout
#define STG_OFF    0                      // 5120 u32 staged keys (async DMA)
#define L_OFF      20480                  // 5120 u64 sort keys
#define REDK_OFF   61440                  // 256 u64
#define REDP_OFF   63488                  // 256 i32
#define SELPOS_OFF 64512                  // 64 i32
#define SELIND_OFF 64768                  // 64 u32
#define SELVAL_OFF 65024                  // 64 f32
#define SMEM2_BYTES 65280

typedef int v4i_t __attribute__((ext_vector_type(4)));
typedef int v8i_t __attribute__((ext_vector_type(8)));

// ---------------------------------------------------------------------------
// Stage 1: one block per (b,c). sigmoid + 3x3 NMS + exact top-64 (radix
// select over float-bit keys). Emits unordered (key, spatial_idx) pairs.
// ---------------------------------------------------------------------------
__global__ void __launch_bounds__(256)
stage1_nms_topk(const float* __restrict__ hm,
                unsigned* __restrict__ outk,
                unsigned* __restrict__ outi)
{
    extern __shared__ char smem[];
    float*    pad  = (float*)smem;
    unsigned* keys = (unsigned*)(smem + KEYS_OFF);
    unsigned* hist = (unsigned*)(smem + HIST_OFF);
    unsigned* ctrl = (unsigned*)(smem + CTRL_OFF);
    // ctrl[0]=chosen prefix, ctrl[2]=krem, ctrl[3]=cnt_hi, ctrl[4]=cnt_tie

    const int tid = threadIdx.x;
    const int bc  = blockIdx.x;                       // b*C + c
    const float* src = hm + (size_t)bc * HW_;

    // ---- fill padded tile with -inf (border cells stay -inf) -------------
    for (int i = tid; i < PFLOATS; i += 256) pad[i] = -INFINITY;
    __syncthreads();

    // ---- TDM: one descriptor DMA loads the plane with pad insertion ------
    {
        const unsigned lds_base = __builtin_amdgcn_groupstaticsize();
        if (tid == 0) {                 // TDM ignores EXEC; issue from wave 0
            const unsigned long long ga = (unsigned long long)src;
            v4i_t g0;
            g0[0] = 1;                                          // count=1
            g0[1] = (int)(lds_base + (PSTRIDE + 4) * 4);        // lds_addr
            g0[2] = (int)(unsigned)(ga & 0xFFFFFFFFull);        // gaddr lo
            g0[3] = (int)(((unsigned)(ga >> 32) & 0x01FFFFFFu)  // gaddr hi
                          | 0x80000000u);                       // type=2
            v8i_t g1;
            g1[0] = (2 << 16)      // data_size = 4B
                  | (1 << 20)      // pad_enable
                  | (6 << 22)      // pad_interval: 128 DWORDs
                  | (7 << 25);     // pad_amount: 8 DWORDs
            g1[1] = (int)(128u << 16);   // tensor_dim0 = 128 (bits 79:48)
            g1[2] = (int)(128u << 16);   // tensor_dim1 = 128 (bits 111:80)
            g1[3] = (int)(128u << 16);   // tile_dim0   = 128 (bits 127:112)
            g1[4] = 128;                 // tile_dim1 = 128, tile_dim2 = 0
            g1[5] = 128;                 // tensor_dim0_stride = 128
            g1[6] = (int)(16384u << 16); // tensor_dim1_stride (unused, 2D)
            g1[7] = 0;
            asm volatile("tensor_load_to_lds %0, %1"
                         :: "s"(g0), "s"(g1) : "memory");
            __builtin_amdgcn_s_wait_tensorcnt(0);
        }
    }
    __syncthreads();

    // ---- keys: sigmoid(x) bits if local max else 0 ------------------------
    // sigmoid is strictly monotone, so (x >= max of 8 neighbors) is exactly
    // the reference's hmax==hm condition; suppressed values become +0.0,
    // matching hm*(1-ceil(|hmax-hm|)). Non-negative floats order as uints.
    for (int i = 0; i < HW_ / 256; ++i) {
        const int p = tid + i * 256;
        const int r = p >> 7, c = p & (W_ - 1);
        const int ctr = (r + 1) * PSTRIDE + 4 + c;
        const float x = pad[ctr];
        float m = fmaxf(fmaxf(pad[ctr - PSTRIDE - 1], pad[ctr - PSTRIDE]),
                        pad[ctr - PSTRIDE + 1]);
        m = fmaxf(m, fmaxf(pad[ctr - 1], pad[ctr + 1]));
        m = fmaxf(m, fmaxf(fmaxf(pad[ctr + PSTRIDE - 1], pad[ctr + PSTRIDE]),
                           pad[ctr + PSTRIDE + 1]));
        const float v = (x >= m) ? (1.0f / (1.0f + expf(-x))) : 0.0f;
        keys[p] = __float_as_uint(v);
    }
    __syncthreads();

    // ---- exact radix select: find 64th-largest key ------------------------
    unsigned prefix = 0u, mask = 0u;
    int krem = K_;
    for (int shift = 24; shift >= 0; shift -= 8) {
        hist[tid] = 0u;
        __syncthreads();
        for (int i = 0; i < HW_ / 256; ++i) {
            const unsigned k = keys[tid + i * 256];
            if ((k & mask) == prefix)
                atomicAdd(&hist[(k >> shift) & 255u], 1u);
        }
        __syncthreads();
        if (tid == 0) {
            unsigned cum = 0u;
            int b = 255;
            for (; b > 0; --b) {
                cum += hist[b];
                if ((int)cum >= krem) break;
            }
            if ((int)cum < krem) cum += hist[0];       // b==0 fallthrough
            ctrl[0] = prefix | ((unsigned)b << shift);
            ctrl[2] = (unsigned)(krem - (int)(cum - hist[b]));
        }
        __syncthreads();
        prefix = ctrl[0];
        krem   = (int)ctrl[2];
        mask  |= (0xFFu << shift);
        __syncthreads();
    }
    const unsigned T   = prefix;        // threshold key (krem ties needed)
    const int      nhi = K_ - krem;     // exactly nhi keys are > T

    if (tid == 0) { ctrl[3] = 0u; ctrl[4] = 0u; }
    __syncthreads();

    // ---- emit the top-64 set (unordered; stage 2 sorts globally) ----------
    unsigned* ok = outk + (size_t)bc * K_;
    unsigned* oi = outi + (size_t)bc * K_;
    for (int i = 0; i < HW_ / 256; ++i) {
        const int p = tid + i * 256;
        const unsigned k = keys[p];
        if (k > T) {
            const unsigned s = atomicAdd(&ctrl[3], 1u);
            ok[s] = k; oi[s] = (unsigned)p;
        } else if (k == T) {
            const unsigned s = atomicAdd(&ctrl[4], 1u);
            if ((int)s < krem) { ok[nhi + s] = k; oi[nhi + s] = (unsigned)p; }
        }
    }
}

// ---------------------------------------------------------------------------
// Stage 2: one block per batch. Sorted global top-64 over the 5120
// candidates (64-bit keys -> no ties), then bbox decode + conv_w gather.
// Candidate keys are staged into LDS with the ASYNCcnt DMA path.
// ---------------------------------------------------------------------------
__global__ void __launch_bounds__(256)
stage2_select_decode(const unsigned* __restrict__ s1k,
                     const unsigned* __restrict__ s1i,
                     const float* __restrict__ reg,
                     const float* __restrict__ wh,
                     const float* __restrict__ convw,
                     float* __restrict__ out_bbox,
                     float* __restrict__ out_conv)
{
    extern __shared__ char smem[];
    unsigned*           stage  = (unsigned*)(smem + STG_OFF);
    unsigned long long* L      = (unsigned long long*)(smem + L_OFF);
    unsigned long long* redk   = (unsigned long long*)(smem + REDK_OFF);
    int*                redp   = (int*)(smem + REDP_OFF);
    int*                selpos = (int*)(smem + SELPOS_OFF);
    unsigned*           selind = (unsigned*)(smem + SELIND_OFF);
    float*              selval = (float*)(smem + SELVAL_OFF);

    const int tid = threadIdx.x;
    const int b   = blockIdx.x;
    const size_t cbase = (size_t)b * CK_;

    // ---- async DMA the 5120 candidate keys into LDS -----------------------
    {
        const unsigned lds_base = __builtin_amdgcn_groupstaticsize();
        const unsigned long long sbase = (unsigned long long)(s1k + cbase);
#pragma unroll
        for (int i = 0; i < CK_ / (4 * 256); ++i) {          // 5 x b128 each
            const int q = tid + i * 256;                     // float4 id
            const unsigned lds_addr = lds_base + STG_OFF + (unsigned)(q << 4);
            const unsigned goff = (unsigned)(q << 4);
            asm volatile("global_load_async_to_lds_b128 %0, %1, %2"
                         :: "v"(lds_addr), "v"(goff), "s"(sbase)
                         : "memory");
        }
        asm volatile("s_wait_asynccnt 0" ::: "memory");
    }
    __syncthreads();

    for (int i = tid; i < CK_; i += 256) {
        const unsigned long long key = (unsigned long long)stage[i];
        L[i] = (key << 32) | (unsigned)(CK_ - 1 - i);   // unique low bits
    }
    __syncthreads();

    for (int d = 0; d < K_; ++d) {
        unsigned long long best = 0ull;
        int bp = 0;
        for (int i = tid; i < CK_; i += 256) {
            const unsigned long long v = L[i];
            if (v > best) { best = v; bp = i; }
        }
        redk[tid] = best; redp[tid] = bp;
        __syncthreads();
        for (int s = 128; s > 0; s >>= 1) {
            if (tid < s && redk[tid + s] > redk[tid]) {
                redk[tid] = redk[tid + s]; redp[tid] = redp[tid + s];
            }
            __syncthreads();
        }
        if (tid == 0) {
            const int p = redp[0];
            selpos[d] = p;
            selval[d] = __uint_as_float((unsigned)(redk[0] >> 32));
            L[p] = 0ull;                                 // remove winner
        }
        __syncthreads();
    }

    // ---- bbox decode ------------------------------------------------------
    if (tid < K_) {
        const int d   = tid;
        const int pos = selpos[d];
        const unsigned ind = s1i[cbase + pos];
        selind[d] = ind;
        const float cls = (float)(pos >> 6);             // pos / K
        const float ys  = (float)(ind >> 7);             // ind / W
        const float xs  = (float)(ind & (W_ - 1));       // ind % W
        const size_t pb = (size_t)b * 2 * HW_;
        const float rx = reg[pb + ind];
        const float ry = reg[pb + HW_ + ind];
        const float ww = wh[pb + ind];
        const float hh = wh[pb + HW_ + ind];
        const float cx = xs + rx, cy = ys + ry;
        float* o = out_bbox + ((size_t)b * K_ + d) * 6;
        o[0] = cx - ww * 0.5f;
        o[1] = cy - hh * 0.5f;
        o[2] = cx + ww * 0.5f;
        o[3] = cy + hh * 0.5f;
        o[4] = selval[d];
        o[5] = cls;
    }
    __syncthreads();

    // ---- conv_weight gather: [b,d,j] = convw[b, j, ind[d]] ----------------
    for (int t = tid; t < K_ * 169; t += 256) {
        const int d = t / 169;
        const int j = t - d * 169;
        out_conv[((size_t)b * K_ + d) * 169 + j] =
            convw[((size_t)b * 169 + j) * HW_ + selind[d]];
    }
}

// ---------------------------------------------------------------------------
// seg_feat pass-through copy (8.4MB), b128 vectorized.
// ---------------------------------------------------------------------------
__global__ void __launch_bounds__(256)
seg_copy(const float4* __restrict__ src, float4* __restrict__ dst, int n)
{
    const int i = blockIdx.x * 256 + threadIdx.x;
    if (i < n) dst[i] = src[i];
}

// ---------------------------------------------------------------------------
extern "C" void kernel_launch(void* const* d_in, const int* in_sizes, int n_in,
                              void* d_out, int out_size, void* d_ws, size_t ws_size,
                              hipStream_t stream)
{
    const float* hm    = (const float*)d_in[0];   // [16,80,128,128]
    const float* reg   = (const float*)d_in[1];   // [16,2,128,128]
    const float* wh    = (const float*)d_in[2];   // [16,2,128,128]
    const float* seg   = (const float*)d_in[3];   // [16,8,128,128]
    const float* convw = (const float*)d_in[4];   // [16,169,128,128]

    float* out       = (float*)d_out;
    float* out_bbox  = out;                                   // 16*64*6   = 6144
    float* out_seg   = out + B_ * K_ * 6;                     // 16*8*HW   = 2097152
    float* out_conv  = out_seg + (size_t)B_ * 8 * HW_;        // 16*64*169 = 173056

    unsigned* s1k = (unsigned*)d_ws;                          // [B*C*K]
    unsigned* s1i = s1k + (size_t)B_ * C_ * K_;               // [B*C*K]

    stage1_nms_topk<<<dim3(B_ * C_), dim3(256), SMEM1_BYTES, stream>>>(hm, s1k, s1i);
    stage2_select_decode<<<dim3(B_), dim3(256), SMEM2_BYTES, stream>>>(s1k, s1i, reg, wh,
                                                                       convw, out_bbox, out_conv);
    const int n4 = (B_ * 8 * HW_) / 4;                        // 524288 float4
    seg_copy<<<dim3(n4 / 256), dim3(256), 0, stream>>>((const float4*)seg,
                                                       (float4*)out_seg, n4);
}